// equ_pool_layer_21603685499530
// MI455X (gfx1250) — compile-verified
//
#include <hip/hip_runtime.h>
#include <cfloat>

#define BS   8
#define V    4096
#define C    64
#define A    12
#define KNN  4
#define POOL 1024

typedef __attribute__((ext_vector_type(2))) float v2f;
typedef __attribute__((ext_vector_type(8))) float v8f;

// ---------- lexicographic (dist, index) comparator: used only in final merge ----------
__device__ __forceinline__ bool lt_pair(float d1, int i1, float d2, int i2) {
    return (d1 < d2) || ((d1 == d2) && (i1 < i2));
}

__device__ __forceinline__ void cswap(float& da, int& ia, float& db, int& ib) {
    bool sw = lt_pair(db, ib, da, ia);
    float td = sw ? db : da;  int ti = sw ? ib : ia;
    db = sw ? da : db;        ib = sw ? ia : ib;
    da = td;                  ia = ti;
}

// ---------- branchless sorted-top4 insert (strict <, so ties keep the earlier,
// i.e. smaller, index -- matching jax.lax.top_k tie-breaking within a lane) ----------
__device__ __forceinline__ void insert4_fast(float bd[4], int bi[4], float d, int i) {
    bool c0 = d < bd[0];
    bool c1 = d < bd[1];
    bool c2 = d < bd[2];
    bool c3 = d < bd[3];
    float n3 = c3 ? (c2 ? bd[2] : d) : bd[3];
    int   m3 = c3 ? (c2 ? bi[2] : i) : bi[3];
    float n2 = c2 ? (c1 ? bd[1] : d) : bd[2];
    int   m2 = c2 ? (c1 ? bi[1] : i) : bi[2];
    float n1 = c1 ? (c0 ? bd[0] : d) : bd[1];
    int   m1 = c1 ? (c0 ? bi[0] : i) : bi[1];
    bd[0] = c0 ? d : bd[0];  bi[0] = c0 ? i : bi[0];
    bd[1] = n1;  bi[1] = m1;
    bd[2] = n2;  bi[2] = m2;
    bd[3] = n3;  bi[3] = m3;
}

// ---------- kernel 1: vertices_pool gather ----------
__global__ void vgather_kernel(const float* __restrict__ verts,
                               const int* __restrict__ sidx,
                               float* __restrict__ outv) {
    int t = blockIdx.x * blockDim.x + threadIdx.x;  // B*POOL
    if (t < BS * POOL) {
        int b = t / POOL, p = t % POOL;
        const float* src = verts + ((size_t)b * V + sidx[p]) * 3;
        float* dst = outv + (size_t)t * 3;
        dst[0] = src[0]; dst[1] = src[1]; dst[2] = src[2];
    }
}

// ---------- kernel 2: WMMA kNN top-4 for sampled vertices ----------
// One wave32 per (batch, 16 queries). K=4 trick:
//   A row (candidate m) = (x, y, z, |c|^2),  B col (query n) = (-2qx, -2qy, -2qz, 1)
//   => D[m][n] = |c|^2 - 2<c,q>  == ranking score directly (|q|^2 const per query).
__global__ __launch_bounds__(32) void knn_topk_kernel(const float* __restrict__ verts,
                                                      const int* __restrict__ sidx,
                                                      int* __restrict__ nbr) {
    const int lane = threadIdx.x;      // 0..31, wave32
    const int n    = lane & 15;        // query column within tile
    const int hi   = lane >> 4;        // half select (K 0,1 vs K 2,3 / rows 0-7 vs 8-15)
    const int b    = blockIdx.y;
    const int p    = blockIdx.x * 16 + n;

    const float* vb = verts + (size_t)b * (V * 3);

    const int qidx = sidx[p];
    const float qx = vb[3 * qidx + 0];
    const float qy = vb[3 * qidx + 1];
    const float qz = vb[3 * qidx + 2];

    // B operand (4x16 f32): lanes 0-15 hold K={0,1}, lanes 16-31 hold K={2,3}.
    v2f bop;
    bop.x = hi ? (-2.0f * qz) : (-2.0f * qx);
    bop.y = hi ? 1.0f          : (-2.0f * qy);

    float bd[4] = {FLT_MAX, FLT_MAX, FLT_MAX, FLT_MAX};
    int   bi[4] = {0x7FFFFFFF, 0x7FFFFFFF, 0x7FFFFFFF, 0x7FFFFFFF};

#pragma unroll 2
    for (int base = 0; base < V; base += 16) {
        // Unconditional coord loads (no exec-masked load), then select.
        const float* cv = vb + (size_t)(base + n) * 3;
        float cx = cv[0], cy = cv[1], cz = cv[2];
        float c2 = cx * cx + cy * cy + cz * cz;

        // A operand (16x4 f32): lane m=n holds K={0,1} lo-half / K={2,3} hi-half.
        v2f aop;
        aop.x = hi ? cz : cx;
        aop.y = hi ? c2 : cy;

        v8f acc = {0.0f, 0.0f, 0.0f, 0.0f, 0.0f, 0.0f, 0.0f, 0.0f};
        v8f dm = __builtin_amdgcn_wmma_f32_16x16x4_f32(
            false, aop, false, bop, (short)0, acc, false, false);

        const int rowbase = base + hi * 8;  // this lane's candidate rows: r + 8*hi
#pragma unroll
        for (int r = 0; r < 8; ++r) {
            const int cidx = rowbase + r;
            float score = (cidx == qidx) ? FLT_MAX : dm[r];  // self-exclusion, no branch
            insert4_fast(bd, bi, score, cidx);
        }
    }

    // Cross-half merge: partner lane ^ 16 holds the other 8 row-residues.
    float od[4]; int oi[4];
#pragma unroll
    for (int k = 0; k < 4; ++k) {
        od[k] = __shfl_xor(bd[k], 16, 32);
        oi[k] = __shfl_xor(bi[k], 16, 32);
    }
    // Bitonic 4+4 merge: lows of (asc A, desc B) butterflies are the global top-4.
    cswap(bd[0], bi[0], od[3], oi[3]);
    cswap(bd[1], bi[1], od[2], oi[2]);
    cswap(bd[2], bi[2], od[1], oi[1]);
    cswap(bd[3], bi[3], od[0], oi[0]);
    cswap(bd[0], bi[0], bd[2], bi[2]);
    cswap(bd[1], bi[1], bd[3], bi[3]);
    cswap(bd[0], bi[0], bd[1], bi[1]);
    cswap(bd[2], bi[2], bd[3], bi[3]);

    if (hi == 0) {
        int4 r; r.x = bi[0]; r.y = bi[1]; r.z = bi[2]; r.w = bi[3];
        *(int4*)(nbr + ((size_t)b * POOL + p) * KNN) = r;
    }
}

// ---------- kernel 3: neighbor-feature gather + max-pool over K ----------
__global__ void pool_kernel(const float* __restrict__ fmap,  // (B,C,V,A)
                            const int* __restrict__ nbr,     // (B,POOL,4)
                            float* __restrict__ outf) {      // (B,C,POOL,A)
    int t = blockIdx.x * blockDim.x + threadIdx.x;  // B*C*POOL
    if (t >= BS * C * POOL) return;
    int p  = t % POOL;
    int bc = t / POOL;       // b*C + c
    int b  = bc / C;

    const int4 nb = *(const int4*)(nbr + ((size_t)b * POOL + p) * KNN);
    const float* plane = fmap + (size_t)bc * V * A;

    // A=12 floats per row = 3 x float4 (rows are 48B => 16B aligned).
    const float4* r0 = (const float4*)(plane + (size_t)nb.x * A);
    float4 m0 = r0[0], m1 = r0[1], m2 = r0[2];
    int idxs[3] = {nb.y, nb.z, nb.w};
#pragma unroll
    for (int k = 0; k < 3; ++k) {
        const float4* rk = (const float4*)(plane + (size_t)idxs[k] * A);
        float4 a0 = rk[0], a1 = rk[1], a2 = rk[2];
        m0.x = fmaxf(m0.x, a0.x); m0.y = fmaxf(m0.y, a0.y);
        m0.z = fmaxf(m0.z, a0.z); m0.w = fmaxf(m0.w, a0.w);
        m1.x = fmaxf(m1.x, a1.x); m1.y = fmaxf(m1.y, a1.y);
        m1.z = fmaxf(m1.z, a1.z); m1.w = fmaxf(m1.w, a1.w);
        m2.x = fmaxf(m2.x, a2.x); m2.y = fmaxf(m2.y, a2.y);
        m2.z = fmaxf(m2.z, a2.z); m2.w = fmaxf(m2.w, a2.w);
    }
    float4* dst = (float4*)(outf + (size_t)t * A);
    dst[0] = m0; dst[1] = m1; dst[2] = m2;
}

extern "C" void kernel_launch(void* const* d_in, const int* in_sizes, int n_in,
                              void* d_out, int out_size, void* d_ws, size_t ws_size,
                              hipStream_t stream) {
    const float* verts = (const float*)d_in[0];   // (BS,V,3)
    const float* fmap  = (const float*)d_in[1];   // (BS,C,V,A)
    const int*   sidx  = (const int*)d_in[2];     // (POOL)

    float* outv = (float*)d_out;                  // (BS,POOL,3)
    float* outf = outv + (size_t)BS * POOL * 3;   // (BS,C,POOL,A)

    int* nbr = (int*)d_ws;                        // BS*POOL*4 ints (128 KB)

    vgather_kernel<<<(BS * POOL + 255) / 256, 256, 0, stream>>>(verts, sidx, outv);
    knn_topk_kernel<<<dim3(POOL / 16, BS), 32, 0, stream>>>(verts, sidx, nbr);
    pool_kernel<<<(BS * C * POOL + 255) / 256, 256, 0, stream>>>(fmap, nbr, outf);
}